// Cheb_13116830122345
// MI455X (gfx1250) — compile-verified
//
#include <hip/hip_runtime.h>
#include <math.h>

#define NN   100000
#define EE   1600000
#define GG   512
#define HID  128
#define OUTD 10
#define LAY  5
#define K2   256          // 2*HID
#define BNEPS 1e-5f

typedef __attribute__((ext_vector_type(2))) float v2f;
typedef __attribute__((ext_vector_type(8))) float v8f;

// ---- optional gfx1250 async global->LDS copy ------------------------------
#if defined(__has_builtin)
# if __has_builtin(__builtin_amdgcn_global_load_async_to_lds_b128)
#  define ASYNC_OK 1
# endif
#endif
#ifndef ASYNC_OK
# define ASYNC_OK 0
#endif

#if ASYNC_OK
typedef int v4i __attribute__((vector_size(16)));
typedef __attribute__((address_space(1))) v4i* gas1_t;
typedef __attribute__((address_space(3))) v4i* las3_t;
__device__ __forceinline__ void async_copy16(const void* src, void* dst_lds) {
    __builtin_amdgcn_global_load_async_to_lds_b128((gas1_t)src, (las3_t)dst_lds, 0, 0);
}
__device__ __forceinline__ void async_wait0() {
#if __has_builtin(__builtin_amdgcn_s_wait_asynccnt)
    __builtin_amdgcn_s_wait_asynccnt(0);
#else
    asm volatile("s_wait_asynccnt 0" ::: "memory");
#endif
}
#endif

// ---------------------------------------------------------------- utilities
__global__ void k_zero(float* __restrict__ p, int n) {
    int i  = blockIdx.x * blockDim.x + threadIdx.x;
    int st = gridDim.x * blockDim.x;
    for (; i < n; i += st) p[i] = 0.f;
}

// in-degree via atomics
__global__ void k_deg(const int* __restrict__ dst, float* __restrict__ deg) {
    int i  = blockIdx.x * blockDim.x + threadIdx.x;
    int st = gridDim.x * blockDim.x;
    for (; i < EE; i += st) atomicAdd(&deg[dst[i]], 1.f);
}

// dinv = clip(deg,1)^-0.5  (in place)
__global__ void k_dinv(float* __restrict__ deg) {
    int i = blockIdx.x * blockDim.x + threadIdx.x;
    if (i < NN) deg[i] = rsqrtf(fmaxf(deg[i], 1.f));
}

// acc[dst] += x[src] * dinv[src]   (2 edges per 256-thread block, 128 cols/edge)
__global__ __launch_bounds__(256) void k_scatter(const float* __restrict__ x,
                                                 const float* __restrict__ dinv,
                                                 const int* __restrict__ src,
                                                 const int* __restrict__ dst,
                                                 float* __restrict__ acc) {
    int eo = blockIdx.x * 2 + (threadIdx.x >> 7);
    int c  = threadIdx.x & (HID - 1);
    if (eo < EE) {
        int s = src[eo], d = dst[eo];
        atomicAdd(&acc[(size_t)d * HID + c], x[(size_t)s * HID + c] * dinv[s]);
    }
}

// ------------------------------------------------------ fp32 WMMA GEMM
// hpre[m0:m0+64, :] = [X0 | -dinv*acc] @ W + bias
// block: 256 threads (8 waves); wave w owns cols [16w,16w+16) over 4 M-subtiles.
// A-tile stride 260 floats: 16B-aligned rows, conflict-free column reads.
__global__ __launch_bounds__(256) void k_gemm(const float* __restrict__ x0,
                                              const float* __restrict__ acc,
                                              const float* __restrict__ dinv,
                                              const float* __restrict__ W,    // 256 x 128 row-major
                                              const float* __restrict__ bias, // 128
                                              float* __restrict__ hpre) {
    __shared__ float As[64][K2 + 4];
    __shared__ float sdinv[64];
    const int m0  = blockIdx.x * 64;
    const int tid = threadIdx.x;

    if (tid < 64) {
        int r = m0 + tid;
        sdinv[tid] = (r < NN) ? dinv[r] : 0.f;
    }
    __syncthreads();

    // ---- stage X0 half (cols 0..127): 64 rows x 32 float4 chunks
    for (int idx = tid; idx < 64 * 32; idx += 256) {
        int r   = idx >> 5;
        int c4  = idx & 31;
        int row = m0 + r;
        if (row < NN) {
#if ASYNC_OK
            async_copy16(x0 + (size_t)row * HID + c4 * 4, &As[r][c4 * 4]);
#else
            *(float4*)&As[r][c4 * 4] =
                *(const float4*)(x0 + (size_t)row * HID + c4 * 4);
#endif
        } else {
            float4 z = {0.f, 0.f, 0.f, 0.f};
            *(float4*)&As[r][c4 * 4] = z;
        }
    }
    // ---- stage X1 half (cols 128..255): -dinv * acc  (needs VALU)
    for (int idx = tid; idx < 64 * 32; idx += 256) {
        int r   = idx >> 5;
        int c4  = idx & 31;
        int row = m0 + r;
        float4 v = {0.f, 0.f, 0.f, 0.f};
        if (row < NN) {
            float4 t = *(const float4*)(acc + (size_t)row * HID + c4 * 4);
            float s = -sdinv[r];
            v.x = s * t.x; v.y = s * t.y; v.z = s * t.z; v.w = s * t.w;
        }
        *(float4*)&As[r][HID + c4 * 4] = v;
    }
#if ASYNC_OK
    async_wait0();
#endif
    __syncthreads();

    const int wid  = tid >> 5;
    const int lane = tid & 31;
    const int n0   = wid * 16;
    const int lrow = lane & 15;
    const int kh   = (lane >> 4) * 2;  // lanes 0-15: K=k,k+1 ; lanes 16-31: K=k+2,k+3

    v8f c0 = {}, c1 = {}, c2 = {}, c3 = {};
    #pragma unroll 2
    for (int k = 0; k < K2; k += 4) {
        v2f b;
        b.x = W[(size_t)(k + kh) * HID + n0 + lrow];
        b.y = W[(size_t)(k + kh + 1) * HID + n0 + lrow];
        v2f a0, a1, a2, a3;
        a0.x = As[lrow][k + kh];      a0.y = As[lrow][k + kh + 1];
        a1.x = As[16 + lrow][k + kh]; a1.y = As[16 + lrow][k + kh + 1];
        a2.x = As[32 + lrow][k + kh]; a2.y = As[32 + lrow][k + kh + 1];
        a3.x = As[48 + lrow][k + kh]; a3.y = As[48 + lrow][k + kh + 1];
        c0 = __builtin_amdgcn_wmma_f32_16x16x4_f32(false, a0, false, b, (short)0, c0, false, false);
        c1 = __builtin_amdgcn_wmma_f32_16x16x4_f32(false, a1, false, b, (short)0, c1, false, false);
        c2 = __builtin_amdgcn_wmma_f32_16x16x4_f32(false, a2, false, b, (short)0, c2, false, false);
        c3 = __builtin_amdgcn_wmma_f32_16x16x4_f32(false, a3, false, b, (short)0, c3, false, false);
    }

    // C/D layout: VGPR j -> M = j (lanes 0-15) / j+8 (lanes 16-31); N = lane&15
    const int col   = n0 + lrow;
    const float bv  = bias[col];
    const int rbase = m0 + ((lane >> 4) * 8);
    #pragma unroll
    for (int j = 0; j < 8; ++j) {
        int r = rbase + j;
        if (r < NN)      hpre[(size_t)r * HID + col]        = c0[j] + bv;
        if (r + 16 < NN) hpre[(size_t)(r + 16) * HID + col] = c1[j] + bv;
        if (r + 32 < NN) hpre[(size_t)(r + 32) * HID + col] = c2[j] + bv;
        if (r + 48 < NN) hpre[(size_t)(r + 48) * HID + col] = c3[j] + bv;
    }
}

// ------------------------------------------------------ batch-norm pieces
__global__ __launch_bounds__(HID) void k_stats(const float* __restrict__ hpre,
                                               float* __restrict__ sums,
                                               float* __restrict__ sqs) {
    int c = threadIdx.x;
    float s = 0.f, q = 0.f;
    for (int r = blockIdx.x; r < NN; r += gridDim.x) {
        float v = hpre[(size_t)r * HID + c];
        s += v; q += v * v;
    }
    atomicAdd(&sums[c], s);
    atomicAdd(&sqs[c], q);
}

__global__ __launch_bounds__(HID) void k_bnfin(const float* __restrict__ sums,
                                               const float* __restrict__ sqs,
                                               const float* __restrict__ gamma,
                                               const float* __restrict__ beta,
                                               float* __restrict__ scale,
                                               float* __restrict__ shift) {
    int c = threadIdx.x;
    float mu  = sums[c] * (1.f / NN);
    float var = sqs[c] * (1.f / NN) - mu * mu;
    float sc  = gamma[c] * rsqrtf(var + BNEPS);
    scale[c] = sc;
    shift[c] = beta[c] - mu * sc;
}

// h = relu(hpre*scale+shift); pooled[g] += h  (run-compaction: n2g is sorted)
__global__ __launch_bounds__(HID) void k_apply_pool(const float* __restrict__ hpre,
                                                    const float* __restrict__ scale,
                                                    const float* __restrict__ shift,
                                                    const int* __restrict__ n2g,
                                                    float* __restrict__ hout,
                                                    float* __restrict__ pooled) {
    int c  = threadIdx.x;
    int r0 = blockIdx.x * 64;
    float sc = scale[c], sh = shift[c];
    int cur = -1; float sum = 0.f;
    for (int rr = 0; rr < 64; ++rr) {
        int r = r0 + rr;
        if (r >= NN) break;
        float v = hpre[(size_t)r * HID + c] * sc + sh;
        v = v > 0.f ? v : 0.f;
        hout[(size_t)r * HID + c] = v;
        int g = n2g[r];
        if (g != cur) {
            if (cur >= 0) atomicAdd(&pooled[(size_t)cur * HID + c], sum);
            cur = g; sum = 0.f;
        }
        sum += v;
    }
    if (cur >= 0) atomicAdd(&pooled[(size_t)cur * HID + c], sum);
}

// sum-pool of raw features (layer 0 of hidden_rep)
__global__ __launch_bounds__(HID) void k_pool(const float* __restrict__ x,
                                              const int* __restrict__ n2g,
                                              float* __restrict__ pooled) {
    int c  = threadIdx.x;
    int r0 = blockIdx.x * 64;
    int cur = -1; float sum = 0.f;
    for (int rr = 0; rr < 64; ++rr) {
        int r = r0 + rr;
        if (r >= NN) break;
        float v = x[(size_t)r * HID + c];
        int g = n2g[r];
        if (g != cur) {
            if (cur >= 0) atomicAdd(&pooled[(size_t)cur * HID + c], sum);
            cur = g; sum = 0.f;
        }
        sum += v;
    }
    if (cur >= 0) atomicAdd(&pooled[(size_t)cur * HID + c], sum);
}

// score = sum_i pooled_i @ predW_i + predb_i ; out = [log_softmax(score), mean(pooled[1:])]
__global__ __launch_bounds__(HID) void k_final(const float* __restrict__ pooled, // 6 x G x 128
                                               const float* __restrict__ predW,  // 6 x 128 x 10
                                               const float* __restrict__ predb,  // 6 x 10
                                               float* __restrict__ out) {
    __shared__ float ssc[OUTD];
    __shared__ float sfin[OUTD];
    int g = blockIdx.x, c = threadIdx.x;
    if (c < OUTD) ssc[c] = 0.f;
    __syncthreads();

    float p[LAY + 1];
    #pragma unroll
    for (int i = 0; i <= LAY; ++i)
        p[i] = pooled[((size_t)i * GG + g) * HID + c];

    #pragma unroll
    for (int o = 0; o < OUTD; ++o) {
        float a = 0.f;
        #pragma unroll
        for (int i = 0; i <= LAY; ++i)
            a += p[i] * predW[((size_t)i * HID + c) * OUTD + o];
        atomicAdd(&ssc[o], a);
    }
    // second output: mean of pooled layers 1..5
    out[(size_t)GG * OUTD + (size_t)g * HID + c] =
        (p[1] + p[2] + p[3] + p[4] + p[5]) * 0.2f;
    __syncthreads();

    if (c < OUTD) {
        float b = 0.f;
        #pragma unroll
        for (int i = 0; i <= LAY; ++i) b += predb[i * OUTD + c];
        sfin[c] = ssc[c] + b;
    }
    __syncthreads();
    if (c < OUTD) {
        float m = -1e30f;
        for (int o = 0; o < OUTD; ++o) m = fmaxf(m, sfin[o]);
        float se = 0.f;
        for (int o = 0; o < OUTD; ++o) se += expf(sfin[o] - m);
        out[(size_t)g * OUTD + c] = sfin[c] - m - logf(se);
    }
}

// ---------------------------------------------------------------- driver
extern "C" void kernel_launch(void* const* d_in, const int* in_sizes, int n_in,
                              void* d_out, int out_size, void* d_ws, size_t ws_size,
                              hipStream_t stream) {
    (void)in_sizes; (void)n_in; (void)out_size; (void)ws_size;
    const float* feat  = (const float*)d_in[0];
    const float* chebW = (const float*)d_in[1];   // L x 256 x 128
    const float* chebB = (const float*)d_in[2];   // L x 128
    const float* gamma = (const float*)d_in[3];   // L x 128
    const float* beta  = (const float*)d_in[4];   // L x 128
    const float* predW = (const float*)d_in[5];   // (L+1) x 128 x 10
    const float* predb = (const float*)d_in[6];   // (L+1) x 10
    const int*   src   = (const int*)d_in[7];
    const int*   dst   = (const int*)d_in[8];
    const int*   n2g   = (const int*)d_in[9];
    float* out = (float*)d_out;

    const size_t NH = (size_t)NN * HID;
    float* ws     = (float*)d_ws;
    float* dinv   = ws;                 // NN  (padded slot 100352)
    float* hcur   = ws + 100352;        // N x 128
    float* acc    = hcur + NH;          // N x 128
    float* hpre   = acc + NH;           // N x 128
    float* sums   = hpre + NH;          // 128
    float* sqs    = sums + 128;         // 128
    float* scale  = sqs + 128;          // 128
    float* shift  = scale + 128;        // 128
    float* pooled = shift + 128;        // 6 x G x 128

    // degree normalization
    k_zero<<<256, 256, 0, stream>>>(dinv, NN);
    k_deg<<<2048, 256, 0, stream>>>(dst, dinv);
    k_dinv<<<(NN + 255) / 256, 256, 0, stream>>>(dinv);

    // pooling accumulators
    k_zero<<<256, 256, 0, stream>>>(pooled, (LAY + 1) * GG * HID);
    k_pool<<<(NN + 63) / 64, HID, 0, stream>>>(feat, n2g, pooled);

    const float* x = feat;
    for (int i = 0; i < LAY; ++i) {
        k_zero<<<4096, 256, 0, stream>>>(acc, (int)NH);
        k_scatter<<<EE / 2, 256, 0, stream>>>(x, dinv, src, dst, acc);
        k_gemm<<<(NN + 63) / 64, 256, 0, stream>>>(x, acc, dinv,
                                                   chebW + (size_t)i * K2 * HID,
                                                   chebB + (size_t)i * HID, hpre);
        k_zero<<<1, 256, 0, stream>>>(sums, 256);  // sums+sqs contiguous
        k_stats<<<512, HID, 0, stream>>>(hpre, sums, sqs);
        k_bnfin<<<1, HID, 0, stream>>>(sums, sqs,
                                       gamma + (size_t)i * HID,
                                       beta + (size_t)i * HID, scale, shift);
        k_apply_pool<<<(NN + 63) / 64, HID, 0, stream>>>(
            hpre, scale, shift, n2g, hcur, pooled + (size_t)(i + 1) * GG * HID);
        x = hcur;
    }

    k_final<<<GG, HID, 0, stream>>>(pooled, predW, predb, out);
}